// TransferSH_48722109005891
// MI455X (gfx1250) — compile-verified
//
#include <hip/hip_runtime.h>

typedef __attribute__((ext_vector_type(2))) float v2f;
typedef __attribute__((ext_vector_type(8))) float v8f;

// ---------------------------------------------------------------------------
// Kernel 1: tiny MLP, runs once on a single wave (32 threads).
//   h = glo(1x16) @ w1(16x32) + b1           <- WMMA f32 16x16x4, 8 issues
//   h = layernorm(h)*ln_w + ln_b ; relu
//   res = (h @ w2(32x12) + b2) * 1e-12
//   affine(3x4) = res + eye(3,4) -> ws[0..11]
//   out_scalar  = mean(|res|)
//
// A-matrix (16x4 f32, ISA 7.12.2): lanes 0-15 hold K={0,1} in VGPR{0,1},
// lanes 16-31 hold K={2,3}. Only row M=0 carries glo; other rows are 0.
// B-matrix (4x16 f32): lanes 0-15 hold K={0,1}, lanes 16-31 hold K={2,3},
// N = lane%16 (mirrors the 64x16 B layout pattern).
// D row M=0 = VGPR0 on lanes 0-15.
// ---------------------------------------------------------------------------
__global__ void __launch_bounds__(32)
mlp_affine_kernel(const float* __restrict__ glo,
                  const float* __restrict__ w1,
                  const float* __restrict__ b1,
                  const float* __restrict__ lnw,
                  const float* __restrict__ lnb,
                  const float* __restrict__ w2,
                  const float* __restrict__ b2,
                  float* __restrict__ ws_affine,
                  float* __restrict__ out_scalar) {
  __shared__ float sh[32];
  __shared__ float sabs[12];

  const int lane = threadIdx.x;      // 0..31, one full wave32
  const int half = lane >> 4;        // 0 -> K pair {0,1}, 1 -> {2,3}
  const int m    = lane & 15;        // M index (A) / N index (B)

  v8f acc0 = {};                     // h[0..15]  accumulator (16x16 D, row 0 used)
  v8f acc1 = {};                     // h[16..31] accumulator

#pragma unroll
  for (int kc = 0; kc < 4; ++kc) {   // K = 16 split into 4 chunks of 4
    const int kb = kc * 4 + half * 2;
    v2f a;
    a.x = (m == 0) ? glo[kb + 0] : 0.0f;   // only row M=0 is live
    a.y = (m == 0) ? glo[kb + 1] : 0.0f;
    v2f bA, bB;                             // N-chunks 0..15 and 16..31
    bA.x = w1[(kb + 0) * 32 + m];
    bA.y = w1[(kb + 1) * 32 + m];
    bB.x = w1[(kb + 0) * 32 + 16 + m];
    bB.y = w1[(kb + 1) * 32 + 16 + m];
    // 8 args: (neg_a, A, neg_b, B, c_mod, C, reuse_a, reuse_b)
    acc0 = __builtin_amdgcn_wmma_f32_16x16x4_f32(false, a, false, bA,
                                                 (short)0, acc0, false, false);
    acc1 = __builtin_amdgcn_wmma_f32_16x16x4_f32(false, a, false, bB,
                                                 (short)0, acc1, false, false);
  }

  // Row M=0 of D lives in VGPR0 (element 0) on lanes 0..15.
  if (lane < 16) {
    sh[lane]      = acc0[0];
    sh[16 + lane] = acc1[0];
  }
  __syncthreads();

  float hv = sh[lane] + b1[lane];

  // LayerNorm over the 32 features = one wave32 butterfly reduction.
  float s = hv;
#pragma unroll
  for (int off = 16; off; off >>= 1) s += __shfl_xor(s, off, 32);
  const float mu = s * (1.0f / 32.0f);
  const float dv = hv - mu;
  float q = dv * dv;
#pragma unroll
  for (int off = 16; off; off >>= 1) q += __shfl_xor(q, off, 32);
  const float var = q * (1.0f / 32.0f);

  float hr = dv * rsqrtf(var + 1e-5f) * lnw[lane] + lnb[lane];
  hr = fmaxf(hr, 0.0f);

  __syncthreads();          // done reading old sh contents
  sh[lane] = hr;
  __syncthreads();

  if (lane < 12) {
    float a = b2[lane];
#pragma unroll
    for (int l = 0; l < 32; ++l) a = fmaf(sh[l], w2[l * 12 + lane], a);
    const float res = a * 1e-12f;
    const int r = lane >> 2, c = lane & 3;
    ws_affine[lane] = res + ((r == c) ? 1.0f : 0.0f);   // + eye(3,4)
    sabs[lane] = fabsf(res);
  }
  __syncthreads();

  if (lane == 0) {
    float t = 0.0f;
#pragma unroll
    for (int j = 0; j < 12; ++j) t += sabs[j];
    *out_scalar = t * (1.0f / 12.0f);
  }
}

// ---------------------------------------------------------------------------
// Kernel 2: bandwidth-bound streaming/gather kernel, one point per thread.
//   ~220 B of HBM traffic per point -> ~19 us floor at 23.3 TB/s.
// ---------------------------------------------------------------------------
__global__ void __launch_bounds__(256)
transfer_sh_kernel(const float* __restrict__ positions,
                   const int*   __restrict__ indexes,
                   const float* __restrict__ cam_pos,
                   const float* __restrict__ base_sh,    // (P,3,1)
                   const float* __restrict__ higher_sh,  // (P,3,15)
                   const float* __restrict__ affine,     // 12 floats in ws
                   float* __restrict__ out, int N) {
  const int i = blockIdx.x * blockDim.x + threadIdx.x;
  if (i >= N) return;

  float x = positions[3 * i + 0] - cam_pos[0];
  float y = positions[3 * i + 1] - cam_pos[1];
  float z = positions[3 * i + 2] - cam_pos[2];
  const float rn = rsqrtf(x * x + y * y + z * z);
  x *= rn; y *= rn; z *= rn;

  const float xx = x * x, yy = y * y, zz = z * z;
  const float xy = x * y, yz = y * z, xz = x * z;

  float b[16];
  b[0]  = 0.28209479177387814f;
  b[1]  = -0.4886025119029199f * y;
  b[2]  =  0.4886025119029199f * z;
  b[3]  = -0.4886025119029199f * x;
  b[4]  =  1.0925484305920792f * xy;
  b[5]  = -1.0925484305920792f * yz;
  b[6]  =  0.31539156525252005f * (2.0f * zz - xx - yy);
  b[7]  = -1.0925484305920792f * xz;
  b[8]  =  0.5462742152960396f * (xx - yy);
  b[9]  = -0.5900435899266435f * y * (3.0f * xx - yy);
  b[10] =  2.890611442640554f  * xy * z;
  b[11] = -0.4570457994644658f * y * (4.0f * zz - xx - yy);
  b[12] =  0.3731763325901154f * z * (2.0f * zz - 3.0f * xx - 3.0f * yy);
  b[13] = -0.4570457994644658f * x * (4.0f * zz - xx - yy);
  b[14] =  1.445305721320277f  * z * (xx - yy);
  b[15] = -0.5900435899266435f * x * (xx - yy);

  const long idx = (long)indexes[i];
  const float* __restrict__ bp = base_sh   + idx * 3;    // 3 floats
  const float* __restrict__ hp = higher_sh + idx * 45;   // 45 floats

  float c4[4];
#pragma unroll
  for (int ch = 0; ch < 3; ++ch) {
    float acc = bp[ch] * b[0];
#pragma unroll
    for (int k = 0; k < 15; ++k)
      acc = fmaf(hp[ch * 15 + k], b[k + 1], acc);
    c4[ch] = acc + 0.5f;
  }
  c4[3] = 1.0f;

#pragma unroll
  for (int r = 0; r < 3; ++r) {
    float o = affine[r * 4 + 3];                     // * colors4[3] == 1
    o = fmaf(affine[r * 4 + 0], c4[0], o);
    o = fmaf(affine[r * 4 + 1], c4[1], o);
    o = fmaf(affine[r * 4 + 2], c4[2], o);
    out[3 * i + r] = fminf(fmaxf(o, 0.0f), 1.0f);    // coalesced 12B/lane
  }
}

extern "C" void kernel_launch(void* const* d_in, const int* in_sizes, int n_in,
                              void* d_out, int out_size, void* d_ws, size_t ws_size,
                              hipStream_t stream) {
  // setup_inputs() order:
  // 0 positions (N,3)  1 indexes (N)  2 cam_pos (3)  3 glo_feature (1,16)
  // 4 base_sh (P,3,1)  5 higher_sh (P,3,15)
  // 6 w1 (16,32)  7 b1 (32)  8 ln_w (32)  9 ln_b (32)  10 w2 (32,12)  11 b2 (12)
  const float* positions = (const float*)d_in[0];
  const int*   indexes   = (const int*)  d_in[1];
  const float* cam_pos   = (const float*)d_in[2];
  const float* glo       = (const float*)d_in[3];
  const float* base_sh   = (const float*)d_in[4];
  const float* higher_sh = (const float*)d_in[5];
  const float* w1        = (const float*)d_in[6];
  const float* b1        = (const float*)d_in[7];
  const float* lnw       = (const float*)d_in[8];
  const float* lnb       = (const float*)d_in[9];
  const float* w2        = (const float*)d_in[10];
  const float* b2        = (const float*)d_in[11];

  const int N = in_sizes[0] / 3;
  float* out      = (float*)d_out;
  float* ws_aff   = (float*)d_ws;              // 12 floats: affine(3,4)
  float* out_scal = out + (size_t)3 * N;       // tuple tail: mean(|affine_res|)

  mlp_affine_kernel<<<1, 32, 0, stream>>>(glo, w1, b1, lnw, lnb, w2, b2,
                                          ws_aff, out_scal);

  const int threads = 256;
  const int blocks  = (N + threads - 1) / threads;
  transfer_sh_kernel<<<blocks, threads, 0, stream>>>(positions, indexes, cam_pos,
                                                     base_sh, higher_sh,
                                                     ws_aff, out, N);
}